// LightGCN_5282809774284
// MI455X (gfx1250) — compile-verified
//
#include <hip/hip_runtime.h>
#include <stdint.h>

// ---------------------------------------------------------------------------
// LightGCN propagation on MI455X (gfx1250).
//
// final = (emb + A*emb + A^2*emb + A^3*emb) / 4,  A sparse COO (4.8M edges),
// emb = [150000 x 64] f32.
//
// Roofline: 1.8 GFLOP total vs ~2.5 GB of L2-resident gather/scatter per
// layer (dense operands are 38.4MB -> fit in 192MB L2). Memory/atomic bound,
// AI ~ 0.25 FLOP/B -> WMMA is not applicable (no dense GEMM present).
// CDNA5 paths used instead: async global->LDS edge-tile streaming
// (global_load_async_to_lds_b128 + s_wait_asynccnt, double buffered) and
// wave32-coalesced gathers (1 x global_load_b64 per lane covers one 256B row).
// ---------------------------------------------------------------------------

#define N_USERS   100000
#define M_ITEMS   50000
#define N_NODES   150000
#define LATENT    64
#define N_ELEMS   (N_NODES * LATENT)          // 9,600,000 floats per buffer
#define TILE      1024                        // edges per LDS tile
#define SPMM_BLOCK 256                        // 8 waves (wave32)
#define SPMM_WAVES (SPMM_BLOCK / 32)
#define EDGES_PER_WAVE (TILE / SPMM_WAVES)    // 128

// ---- CDNA5 async global->LDS copy (ASYNCcnt tracked) ----------------------
__device__ __forceinline__ void async_copy_b128(uint32_t lds_byte_off,
                                                const void* gaddr) {
  // VDST = LDS byte address VGPR, VADDR = 64-bit global address VGPR pair.
  asm volatile("global_load_async_to_lds_b128 %0, %1, off"
               :: "v"(lds_byte_off), "v"(gaddr)
               : "memory");
}
__device__ __forceinline__ void wait_async0() {
  asm volatile("s_wait_asynccnt 0" ::: "memory");
}

// Low 32 bits of a flat pointer into LDS == LDS byte offset (aperture layout).
__device__ __forceinline__ uint32_t lds_off(const void* p) {
  return (uint32_t)(uintptr_t)p;
}

// ---------------------------------------------------------------------------
// SpMM: nxt[r*64 + d] += val * cur[c*64 + d]  for each COO edge (r, c, val).
// Edge tiles are async-streamed into LDS (double buffered). One edge is
// processed by a full wave: lane l handles dims (2l, 2l+1) -> coalesced
// 256B gather per edge, 2 global_atomic_add_f32 per lane (L2-resident).
// ---------------------------------------------------------------------------
__global__ __launch_bounds__(SPMM_BLOCK)
void lightgcn_spmm_kernel(const int* __restrict__ rows,
                          const int* __restrict__ cols,
                          const float* __restrict__ vals,
                          const float* __restrict__ cur,
                          float* __restrict__ nxt,
                          int nnz, int ntiles) {
  __shared__ __align__(16) int   s_row[2][TILE];
  __shared__ __align__(16) int   s_col[2][TILE];
  __shared__ __align__(16) float s_val[2][TILE];

  const int tid  = threadIdx.x;
  const int wave = tid >> 5;
  const int lane = tid & 31;

  auto issue_tile = [&](int t, int buf) {
    // each thread copies 4 edges (16B) of each array; nnz*4 is 16B-divisible
    const int e0 = t * TILE + tid * 4;
    if (e0 < nnz) {
      async_copy_b128(lds_off(&s_row[buf][tid * 4]), rows + e0);
      async_copy_b128(lds_off(&s_col[buf][tid * 4]), cols + e0);
      async_copy_b128(lds_off(&s_val[buf][tid * 4]), vals + e0);
    }
  };

  int t = blockIdx.x;
  if (t >= ntiles) return;

  int buf = 0;
  issue_tile(t, buf);                       // prologue: prefetch first tile

  for (; t < ntiles; t += gridDim.x) {
    wait_async0();                          // my wave's copies for tile t done
    __syncthreads();                        // all waves' copies visible

    const int tn = t + (int)gridDim.x;      // prefetch next tile into other buf
    if (tn < ntiles) issue_tile(tn, buf ^ 1);

    const int ebase = wave * EDGES_PER_WAVE;
    const int gbase = t * TILE + ebase;
#pragma unroll 4
    for (int i = 0; i < EDGES_PER_WAVE; ++i) {
      if (gbase + i >= nnz) break;          // wave-uniform guard
      const int   r = s_row[buf][ebase + i];   // ds_load, broadcast
      const int   c = s_col[buf][ebase + i];
      const float v = s_val[buf][ebase + i];
      const float2 x = *(const float2*)(cur + ((size_t)c << 6) + (lane << 1));
      float* dst = nxt + ((size_t)r << 6) + (lane << 1);
      atomicAdd(dst + 0, v * x.x);          // global_atomic_add_f32 (L2)
      atomicAdd(dst + 1, v * x.y);
    }
    __syncthreads();                        // everyone done reading s_*[buf]
    buf ^= 1;
  }
}

// ---------------------------------------------------------------------------
// Elementwise helpers (float4 vectorized, HBM/L2 streaming).
// ---------------------------------------------------------------------------
__global__ void lightgcn_init_kernel(const float4* __restrict__ ue,
                                     const float4* __restrict__ ie,
                                     float4* __restrict__ acc,
                                     float4* __restrict__ cur) {
  const int i = blockIdx.x * blockDim.x + threadIdx.x;
  const int nu4 = (N_USERS * LATENT) / 4;
  if (i < N_ELEMS / 4) {
    const float4 v = (i < nu4) ? ue[i] : ie[i - nu4];
    acc[i] = v;
    cur[i] = v;
  }
}

__global__ void lightgcn_zero_kernel(float4* __restrict__ p) {
  const int i = blockIdx.x * blockDim.x + threadIdx.x;
  if (i < N_ELEMS / 4) p[i] = make_float4(0.f, 0.f, 0.f, 0.f);
}

__global__ void lightgcn_addscale_kernel(float4* __restrict__ acc,
                                         const float4* __restrict__ nxt,
                                         float s) {
  const int i = blockIdx.x * blockDim.x + threadIdx.x;
  if (i < N_ELEMS / 4) {
    float4 a = acc[i];
    const float4 b = nxt[i];
    a.x = (a.x + b.x) * s;
    a.y = (a.y + b.y) * s;
    a.z = (a.z + b.z) * s;
    a.w = (a.w + b.w) * s;
    acc[i] = a;
  }
}

// ---------------------------------------------------------------------------
extern "C" void kernel_launch(void* const* d_in, const int* in_sizes, int n_in,
                              void* d_out, int out_size, void* d_ws,
                              size_t ws_size, hipStream_t stream) {
  const float* user_emb = (const float*)d_in[0];
  const float* item_emb = (const float*)d_in[1];
  const int*   edge_row = (const int*)d_in[2];
  const int*   edge_col = (const int*)d_in[3];
  const float* edge_val = (const float*)d_in[4];
  const int    nnz      = in_sizes[2];

  float* acc = (float*)d_out;               // running sum of layer outputs
  float* cur = (float*)d_ws;                // ping buffer  (38.4 MB)
  float* nxt = cur + N_ELEMS;               // pong buffer  (38.4 MB)

  const int tpb = 256;
  const int n4  = N_ELEMS / 4;
  const int eb  = (n4 + tpb - 1) / tpb;

  hipLaunchKernelGGL(lightgcn_init_kernel, dim3(eb), dim3(tpb), 0, stream,
                     (const float4*)user_emb, (const float4*)item_emb,
                     (float4*)acc, (float4*)cur);

  const int ntiles = (nnz + TILE - 1) / TILE;
  const int spmm_blocks = ntiles < 1024 ? ntiles : 1024; // persistent, so the
                                                         // double buffer works
  for (int layer = 0; layer < 3; ++layer) {
    hipLaunchKernelGGL(lightgcn_zero_kernel, dim3(eb), dim3(tpb), 0, stream,
                       (float4*)nxt);
    hipLaunchKernelGGL(lightgcn_spmm_kernel, dim3(spmm_blocks),
                       dim3(SPMM_BLOCK), 0, stream,
                       edge_row, edge_col, edge_val, cur, nxt, nnz, ntiles);
    const float s = (layer == 2) ? 0.25f : 1.0f;   // fuse /4 into last add
    hipLaunchKernelGGL(lightgcn_addscale_kernel, dim3(eb), dim3(tpb), 0,
                       stream, (float4*)acc, (const float4*)nxt, s);
    float* tmp = cur; cur = nxt; nxt = tmp;        // ping-pong
  }
}